// Part_Graph_51539607552474
// MI455X (gfx1250) — compile-verified
//
#include <hip/hip_runtime.h>
#include <math.h>

// ---------------------------------------------------------------------------
// Part-graph DCN message passing for MI455X (gfx1250, wave32, WMMA).
//
// Roofline: ~30 GFLOP fp32, ~150 MB live -> compute-bound. The dense
// channel-mix GEMMs of the deformable convs run on V_WMMA_F32_16X16X4_F32
// (full fp32 precision on the matrix pipe):
//   D(16 outs x 16 pixels) += A(16 outs x 4 ch) * B(4 ch x 16 pixels)
// Per 32-pixel wave tile: 9 taps x 5 K-chunks x MT x 2 N-tiles WMMAs.
// Bilinear sampling (data-dependent gather) stays on VALU; per-wave LDS
// marshals lane samples into the wave-striped B layout (ds_load_b64).
// Weight A tiles are zero-padded to MT*16 rows in LDS so the A loads are
// unconditional (no saveexec churn between WMMAs). The per-wave staging
// buffer needs no block barrier: DS ops of a wave complete in order, so the
// compiler's s_wait_dscnt alone sequences store->load within the wave.
// segment_max is fused as uint atomicMax (messages are >= 0).
// ---------------------------------------------------------------------------

#define P_  6
#define HD_ 10
#define E_  12
#define B_  4
#define H_  128
#define W_  128
#define HW_ (H_*W_)

typedef float v2f __attribute__((ext_vector_type(2)));
typedef float v8f __attribute__((ext_vector_type(8)));

static __device__ __forceinline__ v8f wmma_f32_16x16x4(v2f a, v2f b, v8f c) {
  // 8-arg pattern: (neg_a, A, neg_b, B, c_mod, C, reuse_a, reuse_b)
  return __builtin_amdgcn_wmma_f32_16x16x4_f32(false, a, false, b, (short)0, c,
                                               false, false);
}

static __device__ __forceinline__ float sigmoidf_(float x) {
  return 1.f / (1.f + __expf(-x));
}

// ---------------------------------------------------------------------------
// Kernel 1: gate maps.
// dp_att[p,b,y,x]  = sigmoid( xp[p,b,:] . w_att[p] + b_att[p] )
// att_dec[p,b,y,x] = sigmoid( [xh[half(p)]; xp[p]] . w_dec[p] + b_dec[p] )
// ---------------------------------------------------------------------------
__global__ __launch_bounds__(256)
void gates_kernel(const float* __restrict__ xp, const float* __restrict__ xh,
                  const float* __restrict__ w_att, const float* __restrict__ b_att,
                  const float* __restrict__ w_dec, const float* __restrict__ b_dec,
                  float* __restrict__ att_dec_out, float* __restrict__ dp_att_out)
{
  const int pix = blockIdx.x * blockDim.x + threadIdx.x;      // 0..HW-1
  const int pb  = blockIdx.y;                                 // p*B + b
  const int p = pb / B_, b = pb % B_;
  const int half = (p >= 4) ? 1 : 0;                          // PART2HALF
  const float* xpb = xp + (size_t)pb * HD_ * HW_ + pix;
  const float* xhb = xh + (size_t)(half*B_ + b) * HD_ * HW_ + pix;

  float s_att = b_att[p];
  float s_dec = b_dec[p];
  #pragma unroll
  for (int c = 0; c < HD_; ++c) {
    const float vxh = xhb[(size_t)c*HW_];
    const float vxp = xpb[(size_t)c*HW_];
    s_dec = fmaf(w_dec[p*2*HD_ + c],        vxh, s_dec);  // ch 0..9  : xh_sel
    s_dec = fmaf(w_dec[p*2*HD_ + HD_ + c],  vxp, s_dec);  // ch 10..19: xp
    s_att = fmaf(w_att[p*HD_ + c],          vxp, s_att);
  }
  att_dec_out[(size_t)pb*HW_ + pix] = sigmoidf_(s_dec);
  dp_att_out [(size_t)pb*HW_ + pix] = sigmoidf_(s_att);
}

// ---------------------------------------------------------------------------
// Kernel 2/3: one deformable-conv + BN + ReLU layer (templated).
// Block = 256 threads = 8 waves; each lane owns one pixel (32 px / wave).
// IS_L2 additionally applies edge gates and segment-max (atomicMax) into xpp.
// ---------------------------------------------------------------------------
template<int OUTC, bool IS_L2>
__global__ __launch_bounds__(256)
void dcn_kernel(const float* __restrict__ xp,  const float* __restrict__ h_in,
                const int*   __restrict__ src, const int*   __restrict__ dst,
                const float* __restrict__ w_off, const float* __restrict__ b_off,
                const float* __restrict__ w,   const float* __restrict__ sc,
                const float* __restrict__ bi,  const float* __restrict__ dpA,
                float* __restrict__ out, int eBase)
{
  constexpr int MT  = (OUTC + 15) / 16;       // M tiles of 16 output channels
  constexpr int MTR = MT * 16;                // padded A rows (zero-filled)
  __shared__ __align__(16) float s_woff[9*20*27];   // offset conv W, [t][c][o]
  __shared__ __align__(16) float s_w[9*MTR*20];     // main W, [t][o][c], padded
  __shared__ __align__(16) float s_samp[8][32][20]; // per-wave sample staging
  __shared__ float s_scale[OUTC], s_bias[OUTC], s_boff[27];

  const int tid = threadIdx.x;
  const int yb  = blockIdx.y;            // local (edge,batch) index for h bufs
  const int e   = eBase + yb / B_;
  const int b   = yb % B_;
  const int srcP = src[e], dstP = dst[e];

  for (int i = tid; i < 27*20*9; i += 256) {
    const int o = i / 180, c = (i / 9) % 20, t = i % 9;
    s_woff[t*540 + c*27 + o] = w_off[(size_t)e*4860 + i];
  }
  // padded A tile: rows OUTC..MTR-1 are zero so A loads are unconditional
  for (int i = tid; i < 9*MTR*20; i += 256) {
    const int t = i / (MTR*20), o = (i / 20) % MTR, c = i % 20;
    s_w[i] = (o < OUTC) ? w[(size_t)e*(OUTC*180) + o*180 + c*9 + t] : 0.f;
  }
  if (tid < OUTC) { s_scale[tid] = sc[e*OUTC + tid]; s_bias[tid] = bi[e*OUTC + tid]; }
  if (tid < 27)   { s_boff[tid]  = b_off[e*27 + tid]; }
  __syncthreads();

  const int pixBase = blockIdx.x * 256;
  const int pix = pixBase + tid;
  const int y = pix >> 7, x = pix & (W_ - 1);

  const float* inA; const float* inB = nullptr;
  if constexpr (IS_L2) {
    inA = h_in + (size_t)yb * 20 * HW_;
  } else {
    inA = xp + (size_t)(dstP*B_ + b) * HD_ * HW_;   // channels 0..9
    inB = xp + (size_t)(srcP*B_ + b) * HD_ * HW_;   // channels 10..19
  }
  auto load_in = [&](int c, int yy, int xx) -> float {
    if constexpr (IS_L2) {
      return inA[(size_t)c*HW_ + yy*W_ + xx];
    } else {
      return (c < HD_) ? inA[(size_t)c*HW_ + yy*W_ + xx]
                       : inB[(size_t)(c - HD_)*HW_ + yy*W_ + xx];
    }
  };

  // ---- offset/mask conv: 27 outputs per pixel, SAME padding (scalar) ----
  float om[27];
  #pragma unroll
  for (int o = 0; o < 27; ++o) om[o] = s_boff[o];
  for (int t = 0; t < 9; ++t) {
    const int yy = y + t/3 - 1, xx = x + t%3 - 1;
    if (yy >= 0 && yy < H_ && xx >= 0 && xx < W_) {
      for (int c = 0; c < 20; ++c) {
        const float v = load_in(c, yy, xx);
        const float* wrow = &s_woff[t*540 + c*27];
        #pragma unroll
        for (int o = 0; o < 27; ++o) om[o] = fmaf(v, wrow[o], om[o]);
      }
    }
  }

  // ---- WMMA accumulation over 9 deformable taps ----
  const int wave  = tid >> 5, lane = tid & 31;
  const int n     = lane & 15;          // A row / B,D column within tile
  const int kHalf = (lane >> 4) * 2;    // K rows {k,k+1} vs {k+2,k+3}
  const int rAdd  = (lane >> 4) * 8;    // D row offset for lanes 16..31

  v8f acc[MT][2];
  #pragma unroll
  for (int mt = 0; mt < MT; ++mt)
    #pragma unroll
    for (int nt = 0; nt < 2; ++nt) {
      v8f z = {0.f,0.f,0.f,0.f,0.f,0.f,0.f,0.f};
      acc[mt][nt] = z;
    }

  #pragma unroll
  for (int t = 0; t < 9; ++t) {
    const float py = (float)y + (float)(t/3 - 1) + om[2*t];
    const float px = (float)x + (float)(t%3 - 1) + om[2*t + 1];
    const float m  = sigmoidf_(om[18 + t]);
    const float y0f = floorf(py), x0f = floorf(px);
    const float wy = py - y0f, wx = px - x0f;
    const int y0 = (int)y0f, x0 = (int)x0f, y1 = y0 + 1, x1 = x0 + 1;
    const bool vy0 = (unsigned)y0 < (unsigned)H_, vy1 = (unsigned)y1 < (unsigned)H_;
    const bool vx0 = (unsigned)x0 < (unsigned)W_, vx1 = (unsigned)x1 < (unsigned)W_;
    const int cy0 = min(max(y0,0),H_-1), cy1 = min(max(y1,0),H_-1);
    const int cx0 = min(max(x0,0),W_-1), cx1 = min(max(x1,0),W_-1);
    const float f00 = (vy0 && vx0) ? (1.f-wy)*(1.f-wx)*m : 0.f;
    const float f01 = (vy0 && vx1) ? (1.f-wy)*wx*m       : 0.f;
    const float f10 = (vy1 && vx0) ? wy*(1.f-wx)*m       : 0.f;
    const float f11 = (vy1 && vx1) ? wy*wx*m             : 0.f;

    for (int c = 0; c < 20; ++c) {      // masked bilinear sample -> LDS
      const float sv = f00 * load_in(c, cy0, cx0)
                     + f01 * load_in(c, cy0, cx1)
                     + f10 * load_in(c, cy1, cx0)
                     + f11 * load_in(c, cy1, cx1);
      s_samp[wave][lane][c] = sv;
    }
    // No block barrier needed: s_samp slice is private to this wave and a
    // wave's DS ops complete in order; the compiler's s_wait_dscnt sequences
    // the store->load dependency below.

    #pragma unroll
    for (int kc = 0; kc < 5; ++kc) {    // K = 20 channels, 4 at a time
      const int kidx = kc*4 + kHalf;
      #pragma unroll
      for (int nt = 0; nt < 2; ++nt) {  // two 16-pixel N tiles per wave
        const v2f Bv = *(const v2f*)&s_samp[wave][nt*16 + n][kidx];
        #pragma unroll
        for (int mt = 0; mt < MT; ++mt) {
          const v2f Av = *(const v2f*)&s_w[((t*MTR) + mt*16 + n)*20 + kidx];
          acc[mt][nt] = wmma_f32_16x16x4(Av, Bv, acc[mt][nt]);
        }
      }
    }
  }

  // ---- BN scale/bias + ReLU, then store (L1) or gate+segment-max (L2) ----
  #pragma unroll
  for (int nt = 0; nt < 2; ++nt) {
    const int pixL = pixBase + wave*32 + nt*16 + n;
    float g = 0.f;
    if constexpr (IS_L2) {
      const float gs = dpA[(size_t)(srcP*B_ + b)*HW_ + pixL];
      const float gd = dpA[(size_t)(dstP*B_ + b)*HW_ + pixL];
      g = (1.f - gs) * gd;              // a2b gate, >= 0
    }
    #pragma unroll
    for (int mt = 0; mt < MT; ++mt) {
      #pragma unroll
      for (int r = 0; r < 8; ++r) {
        const int o = mt*16 + rAdd + r; // D row layout: M=r (+8 upper lanes)
        if (o < OUTC) {
          const float v = fmaxf(fmaf(acc[mt][nt][r], s_scale[o], s_bias[o]), 0.f);
          if constexpr (IS_L2) {
            // non-negative floats: uint bit order == float order
            atomicMax((unsigned int*)&out[((size_t)(dstP*B_ + b)*OUTC + o)*HW_ + pixL],
                      __float_as_uint(v * g));
          } else {
            out[((size_t)yb*OUTC + o)*HW_ + pixL] = v;
          }
        }
      }
    }
  }
}

// ---------------------------------------------------------------------------
// Kernel 4: xp_new = xp + relu( w_u[p] @ [xp; xpp; att_dec*xh_sel] + b_u[p] )
// ---------------------------------------------------------------------------
__global__ __launch_bounds__(256)
void update_kernel(const float* __restrict__ xp, const float* __restrict__ xh,
                   const float* __restrict__ xpp, const float* __restrict__ att_dec,
                   const float* __restrict__ w_u, const float* __restrict__ b_u,
                   float* __restrict__ xp_new)
{
  __shared__ float s_wu[HD_*3*HD_];
  __shared__ float s_bu[HD_];
  const int pb = blockIdx.y;
  const int p = pb / B_, b = pb % B_;
  const int tid = threadIdx.x;
  for (int i = tid; i < HD_*3*HD_; i += blockDim.x) s_wu[i] = w_u[p*HD_*3*HD_ + i];
  if (tid < HD_) s_bu[tid] = b_u[p*HD_ + tid];
  __syncthreads();

  const int pix  = blockIdx.x * blockDim.x + tid;
  const int half = (p >= 4) ? 1 : 0;
  const float ad = att_dec[(size_t)pb*HW_ + pix];
  const float* xpb  = xp  + (size_t)pb * HD_ * HW_ + pix;
  const float* xppb = xpp + (size_t)pb * HD_ * HW_ + pix;
  const float* xhb  = xh  + (size_t)(half*B_ + b) * HD_ * HW_ + pix;

  float u[3*HD_];
  #pragma unroll
  for (int c = 0; c < HD_; ++c) {
    u[c]         = xpb [(size_t)c*HW_];
    u[HD_ + c]   = xppb[(size_t)c*HW_];
    u[2*HD_ + c] = ad * xhb[(size_t)c*HW_];      // xhp
  }
  #pragma unroll
  for (int d = 0; d < HD_; ++d) {
    float a = s_bu[d];
    #pragma unroll
    for (int c = 0; c < 3*HD_; ++c) a = fmaf(s_wu[d*3*HD_ + c], u[c], a);
    xp_new[(size_t)pb*HD_*HW_ + (size_t)d*HW_ + pix] = u[d] + fmaxf(a, 0.f);
  }
}

// ---------------------------------------------------------------------------
extern "C" void kernel_launch(void* const* d_in, const int* in_sizes, int n_in,
                              void* d_out, int out_size, void* d_ws, size_t ws_size,
                              hipStream_t stream)
{
  (void)in_sizes; (void)n_in; (void)out_size;
  const float* xp     = (const float*)d_in[1];
  const float* xh     = (const float*)d_in[2];
  const int*   src    = (const int*)d_in[3];
  const int*   dst    = (const int*)d_in[4];
  const float* w_off1 = (const float*)d_in[5];
  const float* b_off1 = (const float*)d_in[6];
  const float* w1     = (const float*)d_in[7];
  const float* s1     = (const float*)d_in[8];
  const float* bb1    = (const float*)d_in[9];
  const float* w_off2 = (const float*)d_in[10];
  const float* b_off2 = (const float*)d_in[11];
  const float* w2     = (const float*)d_in[12];
  const float* s2     = (const float*)d_in[13];
  const float* bb2    = (const float*)d_in[14];
  const float* w_att  = (const float*)d_in[15];
  const float* b_att  = (const float*)d_in[16];
  const float* w_dec  = (const float*)d_in[17];
  const float* b_dec  = (const float*)d_in[18];
  const float* w_u    = (const float*)d_in[19];
  const float* b_u    = (const float*)d_in[20];

  float* xp_new  = (float*)d_out;                               // (P,B,10,H,W)
  float* att_dec = xp_new + (size_t)P_*B_*HD_*HW_;              // (P,B,1,H,W)
  float* dp_att  = att_dec + (size_t)P_*B_*HW_;                 // (P,B,1,H,W)

  float* xpp   = (float*)d_ws;                                  // (P,B,10,H,W)
  float* h_buf = xpp + (size_t)P_*B_*HD_*HW_;                   // hidden h

  const size_t need_full =
      ((size_t)P_*B_*HD_ + (size_t)E_*B_*20) * HW_ * sizeof(float);
  const bool full = (ws_size >= need_full);

  hipMemsetAsync(xpp, 0, (size_t)P_*B_*HD_*HW_*sizeof(float), stream);

  const dim3 blk(256);
  gates_kernel<<<dim3(HW_/256, P_*B_), blk, 0, stream>>>(
      xp, xh, w_att, b_att, w_dec, b_dec, att_dec, dp_att);

  if (full) {   // all edges in flight; h_buf holds (E,B,20,H,W)
    dcn_kernel<20, false><<<dim3(HW_/256, E_*B_), blk, 0, stream>>>(
        xp, nullptr, src, dst, w_off1, b_off1, w1, s1, bb1, nullptr, h_buf, 0);
    dcn_kernel<10, true ><<<dim3(HW_/256, E_*B_), blk, 0, stream>>>(
        xp, h_buf, src, dst, w_off2, b_off2, w2, s2, bb2, dp_att, xpp, 0);
  } else {      // serialize edges; h_buf reused as (B,20,H,W) per edge
    for (int e = 0; e < E_; ++e) {
      dcn_kernel<20, false><<<dim3(HW_/256, B_), blk, 0, stream>>>(
          xp, nullptr, src, dst, w_off1, b_off1, w1, s1, bb1, nullptr, h_buf, e);
      dcn_kernel<10, true ><<<dim3(HW_/256, B_), blk, 0, stream>>>(
          xp, h_buf, src, dst, w_off2, b_off2, w2, s2, bb2, dp_att, xpp, e);
    }
  }

  update_kernel<<<dim3(HW_/256, P_*B_), blk, 0, stream>>>(
      xp, xh, xpp, att_dec, w_u, b_u, xp_new);
}